// BPRLossWithNoClick_78417512891010
// MI455X (gfx1250) — compile-verified
//
#include <hip/hip_runtime.h>
#include <stdint.h>

// BPR loss: loss = sum_b -(1/(L_b^2 * S)) * sum_{i<L,j<L,s} log_sigmoid(row_i[lab_j] - row_i[neg_{j,s}])
// Strategy: TDM-stream each needed 80KB row of `output` into LDS (double buffered),
// do all gathers from LDS, wave32 reduction, one f32 atomic per block.

typedef unsigned int u32x4 __attribute__((ext_vector_type(4)));
typedef int          i32x8 __attribute__((ext_vector_type(8)));
typedef int          i32x4 __attribute__((ext_vector_type(4)));

#define ROWS_PER_BLOCK 8
#define BLOCK_THREADS 256

// Issue a TDM load of one row (V contiguous f32) from global into LDS.
// D# layout per CDNA5 ISA ch.8 (2D tensor: tile V x 1, tensor V x 1, stride V).
__device__ __forceinline__ void tdm_load_row(const float* gsrc, uint32_t lds_byte_off, int V) {
  uint64_t ga = (uint64_t)(uintptr_t)gsrc;
  u32x4 g0;
  g0[0] = 1u;                                              // count=1, is_restore=0, gather off
  g0[1] = lds_byte_off;                                    // lds_addr (bytes)
  g0[2] = (uint32_t)ga;                                    // global_addr[31:0]
  g0[3] = (uint32_t)((ga >> 32) & 0x01FFFFFFu)             // global_addr[56:32]
        | (2u << 30);                                      // type = 2 ("image")
  uint32_t v = (uint32_t)V;
  i32x8 g1;
  g1[0] = (int)(2u << 16);                                 // data_size = 2 (4 bytes); no mask/pad/iterate
  g1[1] = (int)((v & 0xFFFFu) << 16);                      // tensor_dim0[15:0]
  g1[2] = (int)(((v >> 16) & 0xFFFFu) | (1u << 16));       // tensor_dim0[31:16] | tensor_dim1=1
  g1[3] = (int)((v & 0xFFFFu) << 16);                      // tensor_dim1 hi=0 | tile_dim0 = V (fits 16b)
  g1[4] = 1;                                               // tile_dim1 = 1, tile_dim2 = 0
  g1[5] = (int)v;                                          // tensor_dim0_stride lo32
  g1[6] = 0;                                               // stride hi | tensor_dim1_stride lo
  g1[7] = 0;
  i32x4 z4 = {0, 0, 0, 0};                                 // groups 2/3 unused (<=2D)
  i32x8 z8 = {0, 0, 0, 0, 0, 0, 0, 0};                     // extra group (clang-23 6-arg form)
  __builtin_amdgcn_tensor_load_to_lds(g0, g1, z4, z4, z8, 0);  // cpol = 0
}

__device__ __forceinline__ float log_sigmoid_stable(float x) {
  // log_sigmoid(x) = min(x,0) - log1p(exp(-|x|))  (numerically stable)
  return fminf(x, 0.f) - log1pf(__expf(-fabsf(x)));
}

__global__ __launch_bounds__(BLOCK_THREADS)
void bpr_loss_kernel(const float* __restrict__ output,
                     const int*   __restrict__ labels,
                     const int*   __restrict__ x_lens,
                     const int*   __restrict__ neg_ids,
                     float*       __restrict__ out,
                     int T, int V, int S) {
  extern __shared__ char smem[];          // [buf0: V f32][buf1: V f32][lab: T i32][neg: T*S i32]
  float* buf0  = (float*)smem;
  float* buf1  = buf0 + V;
  int*   lab_s = (int*)(buf1 + V);
  int*   neg_s = lab_s + T;
  __shared__ float red[BLOCK_THREADS / 32];

  const int b     = blockIdx.y;
  const int chunk = blockIdx.x;
  const int tid   = threadIdx.x;
  const int L     = x_lens[b];

  const int r0 = chunk * ROWS_PER_BLOCK;
  const int r1 = min(r0 + ROWS_PER_BLOCK, L);
  if (r0 >= r1) return;                   // uniform per block: safe early-out

  const uint32_t lds_base = (uint32_t)(uintptr_t)smem;   // generic->LDS: low 32b = LDS offset
  const uint32_t buf_off[2] = { lds_base, lds_base + (uint32_t)V * 4u };

  const float* rows_base = output + (size_t)b * (size_t)T * (size_t)V;

  // Prologue: kick off DMA of first row; stage index arrays into LDS meanwhile.
  if (tid < 32) tdm_load_row(rows_base + (size_t)r0 * V, buf_off[r0 & 1], V);
  for (int j = tid; j < T; j += BLOCK_THREADS)     lab_s[j] = labels[b * T + j];
  for (int j = tid; j < T * S; j += BLOCK_THREADS) neg_s[j] = neg_ids[b * T * S + j];
  if (tid < 32) __builtin_amdgcn_s_wait_tensorcnt(0);
  __syncthreads();

  float acc = 0.f;

  if (S == 1 && L <= BLOCK_THREADS) {
    // Fast path (covers the reference shapes: L <= T = 200 <= 256, S = 1).
    // Each thread owns one j; hoist its two gather indices into registers so
    // the per-row body is just 2 LDS gathers + stable log-sigmoid + add.
    const bool active = (tid < L);
    const int myLab = active ? lab_s[tid] : 0;
    const int myNeg = active ? neg_s[tid] : 0;

    for (int r = r0; r < r1; ++r) {
      if ((r + 1 < r1) && (tid < 32))
        tdm_load_row(rows_base + (size_t)(r + 1) * V, buf_off[(r + 1) & 1], V);

      const float* rowbuf = (r & 1) ? buf1 : buf0;
      if (active) {
        const float p = rowbuf[myLab];
        const float n = rowbuf[myNeg];
        acc += log_sigmoid_stable(p - n);
      }
      __syncthreads();                                     // all done reading rowbuf
      if (tid < 32) __builtin_amdgcn_s_wait_tensorcnt(0);  // next row landed in LDS
      __syncthreads();                                     // publish to all waves
    }
  } else {
    // Generic path: strided over j and s.
    for (int r = r0; r < r1; ++r) {
      if ((r + 1 < r1) && (tid < 32))
        tdm_load_row(rows_base + (size_t)(r + 1) * V, buf_off[(r + 1) & 1], V);

      const float* rowbuf = (r & 1) ? buf1 : buf0;
      for (int j = tid; j < L; j += BLOCK_THREADS) {
        const float p = rowbuf[lab_s[j]];
#pragma unroll 1
        for (int s = 0; s < S; ++s) {
          const float n = rowbuf[neg_s[j * S + s]];
          acc += log_sigmoid_stable(p - n);
        }
      }
      __syncthreads();
      if (tid < 32) __builtin_amdgcn_s_wait_tensorcnt(0);
      __syncthreads();
    }
  }

  // Block reduction: wave32 shuffle tree, then cross-wave via LDS.
  for (int m = 16; m > 0; m >>= 1) acc += __shfl_xor(acc, m, 32);
  const int wave = tid >> 5;
  if ((tid & 31) == 0) red[wave] = acc;
  __syncthreads();
  if (tid == 0) {
    float s = 0.f;
#pragma unroll
    for (int w = 0; w < BLOCK_THREADS / 32; ++w) s += red[w];
    const float Lf = (float)L;
    atomicAdd(out, -s / (Lf * Lf * (float)S));
  }
}

__global__ void bpr_zero_kernel(float* o, int n) {
  int i = blockIdx.x * blockDim.x + threadIdx.x;
  if (i < n) o[i] = 0.f;
}

extern "C" void kernel_launch(void* const* d_in, const int* in_sizes, int n_in,
                              void* d_out, int out_size, void* d_ws, size_t ws_size,
                              hipStream_t stream) {
  // Inputs (setup_inputs order): output f32 [B,T,V], labels i32 [B,T],
  // x_lens i32 [B], uids (unused), neg_ids i32 [B,T,S].
  const float* output  = (const float*)d_in[0];
  const int*   labels  = (const int*)d_in[1];
  const int*   x_lens  = (const int*)d_in[2];
  const int*   neg_ids = (const int*)d_in[4];
  float*       out     = (float*)d_out;

  const int B = in_sizes[2];
  const int T = in_sizes[1] / B;
  const int V = (int)((long long)in_sizes[0] / ((long long)B * (long long)T));
  const int S = in_sizes[4] / (B * T);

  bpr_zero_kernel<<<1, 32, 0, stream>>>(out, out_size);

  const int chunks = (T + ROWS_PER_BLOCK - 1) / ROWS_PER_BLOCK;
  const size_t shmem = (size_t)2 * V * sizeof(float)
                     + (size_t)T * sizeof(int)
                     + (size_t)T * S * sizeof(int);   // ~161.6 KB << 320 KB WGP LDS

  static bool attr_set = false;  // idempotent host-side attribute, not stream state
  if (!attr_set) {
    (void)hipFuncSetAttribute((const void*)bpr_loss_kernel,
                              hipFuncAttributeMaxDynamicSharedMemorySize, (int)shmem);
    attr_set = true;
  }

  dim3 grid(chunks, B);
  bpr_loss_kernel<<<grid, BLOCK_THREADS, shmem, stream>>>(
      output, labels, x_lens, neg_ids, out, T, V, S);
}